// MultiStepDecoder_41300405518523
// MI455X (gfx1250) — compile-verified
//
#include <hip/hip_runtime.h>
#include <hip/hip_bf16.h>
#include <math.h>

// ---------------------------------------------------------------------------
// MultiStepDecoder fused kernel for gfx1250 (MI455X)
// Shapes: B=4,T=8 -> BT=32; P=4096 -> N=131072 rows; C=128; Hg=Wg=64;
// d_four=128; d_in=384; HORIZON=5 -> 10 output cols per row.
// ---------------------------------------------------------------------------

#define BT    32
#define P_PTS 4096
#define NROWS (BT * P_PTS)      // 131072
#define CCH   128
#define GW    64
#define GHW   (GW * GW)         // 4096
#define DIN   384
#define OUTC  10

#define ROWS_PER_BLK 32
#define STR_IN 388              // 388 % 64 == 4 -> conflict-free 64b A loads
#define STR_H  132              // 132 % 64 == 4

typedef __attribute__((ext_vector_type(2))) float v2f;
typedef __attribute__((ext_vector_type(8))) float v8f;

// ---------------------------------------------------------------------------
// Pass 1 (optional): transpose h_grid (bt, C, HW) -> ws (bt, HW, C) so the
// bilinear gather reads 512B-contiguous channel vectors per tap.
// ---------------------------------------------------------------------------
__global__ __launch_bounds__(256) void transpose_chw_to_hwc(
    const float* __restrict__ in, float* __restrict__ out)
{
    __shared__ float tile[32][33];
    const int bt = blockIdx.z;
    const int s0 = blockIdx.x * 32;   // HW tile base
    const int c0 = blockIdx.y * 32;   // channel tile base
    const float* src = in + (size_t)bt * (CCH * GHW);
    float* dst = out + (size_t)bt * (GHW * CCH);

    #pragma unroll
    for (int j = threadIdx.y; j < 32; j += 8)
        tile[j][threadIdx.x] = src[(size_t)(c0 + j) * GHW + (s0 + threadIdx.x)];
    __syncthreads();
    #pragma unroll
    for (int j = threadIdx.y; j < 32; j += 8)
        dst[(size_t)(s0 + j) * CCH + (c0 + threadIdx.x)] = tile[threadIdx.x][j];
}

__device__ inline float softplus_f(float x) {
    // numerically stable: max(x,0) + log1p(exp(-|x|))
    return fmaxf(x, 0.0f) + log1pf(__expf(-fabsf(x)));
}

// ---------------------------------------------------------------------------
// Fused decoder. CHW=true samples from original (C,H,W) layout; CHW=false
// samples from the transposed (HW,C) workspace (fast path).
// Block: 256 threads (8 waves), 32 rows.
// ---------------------------------------------------------------------------
template <bool CHW>
__global__ __launch_bounds__(256) void decoder_kernel(
    const float* __restrict__ hsrc,
    const float* __restrict__ xs, const float* __restrict__ ys,
    const float* __restrict__ fB,
    const float* __restrict__ W0, const float* __restrict__ b0,
    const float* __restrict__ W1, const float* __restrict__ b1,
    const float* __restrict__ Wout, const float* __restrict__ bout,
    float* __restrict__ out)
{
    __shared__ float smem[ROWS_PER_BLK * STR_IN];   // 49,664 B
    float* sIn = smem;        // 32 x 388 (decoder input, live through GEMM1)
    float* sH  = smem;        // 32 x 132 (hidden, aliased after GEMM1)

    const int tid = threadIdx.x;

    // ---------------- Phase A: build decoder_in (sampled ++ sin ++ cos) ----
    {
        const int row  = tid >> 3;       // 0..31
        const int part = tid & 7;        // 0..7
        const int rg   = blockIdx.x * ROWS_PER_BLK + row;
        const float x  = xs[rg];
        const float y  = ys[rg];

        if (part < 4) {
            // bilinear sample, 32 channels per thread
            float gx = fminf(fmaxf(x * 63.0f, 0.0f), 63.0f);
            float gy = fminf(fmaxf(y * 63.0f, 0.0f), 63.0f);
            const int x0 = (int)gx, y0 = (int)gy;
            const int x1 = min(x0 + 1, 63), y1 = min(y0 + 1, 63);
            const float wx1 = gx - (float)x0, wy1 = gy - (float)y0;
            const float wx0 = 1.0f - wx1,     wy0 = 1.0f - wy1;
            const float w00 = wy0 * wx0, w01 = wy0 * wx1;
            const float w10 = wy1 * wx0, w11 = wy1 * wx1;
            const int bt  = rg >> 12;    // P = 4096
            const int i00 = y0 * GW + x0, i01 = y0 * GW + x1;
            const int i10 = y1 * GW + x0, i11 = y1 * GW + x1;
            const int c0  = part * 32;

            if (CHW) {
                const float* base = hsrc + (size_t)bt * (CCH * GHW);
                #pragma unroll 4
                for (int c = c0; c < c0 + 32; ++c) {
                    const float* pc = base + (size_t)c * GHW;
                    sIn[row * STR_IN + c] =
                        w00 * pc[i00] + w01 * pc[i01] + w10 * pc[i10] + w11 * pc[i11];
                }
            } else {
                const float* base = hsrc + (size_t)bt * (GHW * CCH);
                const float* p00 = base + (size_t)i00 * CCH;
                const float* p01 = base + (size_t)i01 * CCH;
                const float* p10 = base + (size_t)i10 * CCH;
                const float* p11 = base + (size_t)i11 * CCH;
                #pragma unroll
                for (int c = c0; c < c0 + 32; c += 4) {
                    float4 a = *(const float4*)(p00 + c);
                    float4 b = *(const float4*)(p01 + c);
                    float4 g = *(const float4*)(p10 + c);
                    float4 d = *(const float4*)(p11 + c);
                    float* o = sIn + row * STR_IN + c;
                    o[0] = w00 * a.x + w01 * b.x + w10 * g.x + w11 * d.x;
                    o[1] = w00 * a.y + w01 * b.y + w10 * g.y + w11 * d.y;
                    o[2] = w00 * a.z + w01 * b.z + w10 * g.z + w11 * d.z;
                    o[3] = w00 * a.w + w01 * b.w + w10 * g.w + w11 * d.w;
                }
            }
        } else {
            // Fourier features, 32 freqs per thread
            const int f0 = (part - 4) * 32;
            const float tx = 6.28318530717958647692f * x;
            const float ty = 6.28318530717958647692f * y;
            #pragma unroll 4
            for (int f = f0; f < f0 + 32; ++f) {
                const float pr = tx * fB[f] + ty * fB[128 + f];
                float sv, cv;
                __sincosf(pr, &sv, &cv);
                sIn[row * STR_IN + 128 + f] = sv;   // sin block
                sIn[row * STR_IN + 256 + f] = cv;   // cos block
            }
        }
    }
    __syncthreads();

    // ---------------- WMMA fragment coordinates (wave32) -------------------
    const int lane  = tid & 31;
    const int wave  = tid >> 5;          // 8 waves == 8 N-tiles of 16 cols
    const int l15   = lane & 15;
    const int khalf = (lane >> 4) * 2;   // lanes 16..31 hold K+2,K+3
    const int rofs  = (lane >> 4) * 8;   // D rows: v (+8 for upper lanes)
    const int col   = wave * 16 + l15;   // output column of this lane

    const v8f zero8 = {0.f, 0.f, 0.f, 0.f, 0.f, 0.f, 0.f, 0.f};

    // ---------------- GEMM1: (32 x 384) @ W0 (384 x 128), ReLU -------------
    v8f acc0 = zero8, acc1 = zero8;      // M-tile 0 (rows 0..15), 1 (16..31)
    #pragma unroll 4
    for (int k = 0; k < DIN / 4; ++k) {
        const int kk = 4 * k + khalf;
        v2f a0 = *(const v2f*)(sIn + l15 * STR_IN + kk);
        v2f a1 = *(const v2f*)(sIn + (16 + l15) * STR_IN + kk);
        v2f bv;
        bv.x = W0[kk * CCH + col];
        bv.y = W0[(kk + 1) * CCH + col];
        acc0 = __builtin_amdgcn_wmma_f32_16x16x4_f32(false, a0, false, bv,
                                                     (short)0, acc0, false, false);
        acc1 = __builtin_amdgcn_wmma_f32_16x16x4_f32(false, a1, false, bv,
                                                     (short)0, acc1, false, false);
    }
    __syncthreads();                     // sIn dead -> safe to alias sH
    {
        const float bias = b0[col];
        #pragma unroll
        for (int v = 0; v < 8; ++v)
            sH[(v + rofs) * STR_H + col] = fmaxf(acc0[v] + bias, 0.0f);
        #pragma unroll
        for (int v = 0; v < 8; ++v)
            sH[(16 + v + rofs) * STR_H + col] = fmaxf(acc1[v] + bias, 0.0f);
    }
    __syncthreads();

    // ---------------- GEMM2: (32 x 128) @ W1 (128 x 128), ReLU -------------
    acc0 = zero8; acc1 = zero8;
    #pragma unroll 4
    for (int k = 0; k < CCH / 4; ++k) {
        const int kk = 4 * k + khalf;
        v2f a0 = *(const v2f*)(sH + l15 * STR_H + kk);
        v2f a1 = *(const v2f*)(sH + (16 + l15) * STR_H + kk);
        v2f bv;
        bv.x = W1[kk * CCH + col];
        bv.y = W1[(kk + 1) * CCH + col];
        acc0 = __builtin_amdgcn_wmma_f32_16x16x4_f32(false, a0, false, bv,
                                                     (short)0, acc0, false, false);
        acc1 = __builtin_amdgcn_wmma_f32_16x16x4_f32(false, a1, false, bv,
                                                     (short)0, acc1, false, false);
    }
    __syncthreads();                     // everyone done reading h1
    {
        const float bias = b1[col];
        #pragma unroll
        for (int v = 0; v < 8; ++v)
            sH[(v + rofs) * STR_H + col] = fmaxf(acc0[v] + bias, 0.0f);
        #pragma unroll
        for (int v = 0; v < 8; ++v)
            sH[(16 + v + rofs) * STR_H + col] = fmaxf(acc1[v] + bias, 0.0f);
    }
    __syncthreads();

    // ---------------- GEMM3: (32 x 128) @ Wout (128 x 10) + epilogue -------
    if (wave < 2) {                       // wave == M-tile; N-tile single
        const int m = wave;
        const bool valid = (l15 < OUTC);
        v8f acc = zero8;
        #pragma unroll 4
        for (int k = 0; k < CCH / 4; ++k) {
            const int kk = 4 * k + khalf;
            v2f a = *(const v2f*)(sH + (m * 16 + l15) * STR_H + kk);
            v2f bv;
            bv.x = valid ? Wout[kk * OUTC + l15] : 0.0f;
            bv.y = valid ? Wout[(kk + 1) * OUTC + l15] : 0.0f;
            acc = __builtin_amdgcn_wmma_f32_16x16x4_f32(false, a, false, bv,
                                                        (short)0, acc, false, false);
        }
        if (valid) {
            const float bo = bout[l15];
            #pragma unroll
            for (int v = 0; v < 8; ++v) {
                const int r = m * 16 + v + rofs;
                float val = acc[v] + bo;
                if (l15 & 1) val = softplus_f(val) + 0.01f;   // sigma cols
                out[(size_t)(blockIdx.x * ROWS_PER_BLK + r) * OUTC + l15] = val;
            }
        }
    }
}

// ---------------------------------------------------------------------------
// Host launcher
// ---------------------------------------------------------------------------
extern "C" void kernel_launch(void* const* d_in, const int* in_sizes, int n_in,
                              void* d_out, int out_size, void* d_ws, size_t ws_size,
                              hipStream_t stream)
{
    const float* h_grid = (const float*)d_in[0];
    const float* xs     = (const float*)d_in[1];
    const float* ys     = (const float*)d_in[2];
    const float* fB     = (const float*)d_in[3];
    const float* W0     = (const float*)d_in[4];
    const float* b0     = (const float*)d_in[5];
    const float* W1     = (const float*)d_in[6];
    const float* b1     = (const float*)d_in[7];
    const float* Wout   = (const float*)d_in[8];
    const float* bout   = (const float*)d_in[9];
    float* out = (float*)d_out;

    const size_t transposed_bytes = (size_t)BT * GHW * CCH * sizeof(float); // 64 MB
    const int nblocks = NROWS / ROWS_PER_BLK;   // 4096

    if (ws_size >= transposed_bytes) {
        float* hwc = (float*)d_ws;
        dim3 tb(32, 8);
        dim3 tg(GHW / 32, CCH / 32, BT);        // (128, 4, 32)
        transpose_chw_to_hwc<<<tg, tb, 0, stream>>>(h_grid, hwc);
        decoder_kernel<false><<<nblocks, 256, 0, stream>>>(
            hwc, xs, ys, fB, W0, b0, W1, b1, Wout, bout, out);
    } else {
        decoder_kernel<true><<<nblocks, 256, 0, stream>>>(
            h_grid, xs, ys, fB, W0, b0, W1, b1, Wout, bout, out);
    }
}